// GraphEncoder_89378269430570
// MI455X (gfx1250) — compile-verified
//
#include <hip/hip_runtime.h>
#include <math.h>

typedef float v2f __attribute__((ext_vector_type(2)));
typedef float v8f __attribute__((ext_vector_type(8)));

#define CIO 64          // C_IN == C_OUT == 64
#define BN_EPS 1e-5f

// ---------------- utility atomics ----------------
__device__ __forceinline__ void atomicAddF(float* p, float v) {
  unsafeAtomicAdd(p, v);          // lowers to global_atomic_add_f32 / ds_add_f32
}

__device__ __forceinline__ void atomicMaxF(float* addr, float val) {
  // classic sign-split trick: always lowers to native integer atomics
  if (val >= 0.0f) atomicMax((int*)addr, __float_as_int(val));
  else             atomicMin((unsigned int*)addr, __float_as_uint(val));
}

// ---------------- fill ----------------
__global__ void fill_kernel(float* __restrict__ p, float v, long long n) {
  long long i = (long long)blockIdx.x * blockDim.x + threadIdx.x;
  if (i < n) p[i] = v;
}

// ---------------- degree: deg[col] += w, + self loops ----------------
__global__ void degree_kernel(const int* __restrict__ colA, const float* __restrict__ ew,
                              float* __restrict__ deg, int E, int N) {
  int idx = blockIdx.x * blockDim.x + threadIdx.x;
  if (idx < E)            atomicAddF(&deg[colA[idx]], ew[idx]);
  else if (idx < E + N)   atomicAddF(&deg[idx - E], 1.0f);
}

// ---------------- dinv in place ----------------
__global__ void rsqrt_kernel(float* __restrict__ deg, int N) {
  int i = blockIdx.x * blockDim.x + threadIdx.x;
  if (i < N) { float d = deg[i]; deg[i] = d > 0.0f ? rsqrtf(d) : 0.0f; }
}

// ---------------- one propagation hop: hout[col] += dinv[row]*w*dinv[col] * hin[row] ----
// one wave (32 lanes) per edge; lane handles channels lane and lane+32
__global__ void hop_kernel(const int* __restrict__ rowA, const int* __restrict__ colA,
                           const float* __restrict__ ew, const float* __restrict__ dinv,
                           const float* __restrict__ hin, float* __restrict__ hout,
                           int E, int N) {
  long long gidx = (long long)blockIdx.x * blockDim.x + threadIdx.x;
  int e    = (int)(gidx >> 5);
  int lane = (int)(gidx & 31);
  if (e >= E + N) return;
  int r, c; float w;
  if (e < E) { r = rowA[e]; c = colA[e]; w = ew[e]; }
  else       { r = e - E;   c = r;       w = 1.0f;  }
  float coef = dinv[r] * w * dinv[c];
  const float* s = hin  + (size_t)r * CIO + lane;
  float*       d = hout + (size_t)c * CIO + lane;
  atomicAddF(d,      coef * s[0]);
  atomicAddF(d + 32, coef * s[32]);
}

// ---------------- GEMM (h2 @ W^T + b) -> PReLU -> h3, fused BN-stat accumulation ----
// wave32; each wave: 16 rows x 64 cols via V_WMMA_F32_16X16X4_F32.
// K-outer / tile-inner: 4 independent accumulator chains interleaved so the
// matrix pipe never stalls on the WMMA->WMMA RAW hazard.
__global__ void gemm_bn_kernel(const float* __restrict__ h2, const float* __restrict__ W,
                               const float* __restrict__ bias, const float* __restrict__ slope,
                               float* __restrict__ h3,
                               float* __restrict__ chanSum, float* __restrict__ chanSq,
                               int n) {
  __shared__ float Wl[CIO * CIO];     // 16 KB
  __shared__ float sSum[CIO], sSq[CIO];
  int tid = threadIdx.x;
  for (int i = tid; i < CIO * CIO; i += 256) Wl[i] = W[i];
  if (tid < CIO) { sSum[tid] = 0.0f; sSq[tid] = 0.0f; }
  __syncthreads();

  int wid  = tid >> 5;
  int lane = tid & 31;
  int row0 = (blockIdx.x * 8 + wid) * 16;

  if (row0 < n) {                       // wave-uniform guard -> EXEC all ones inside
    int m  = lane & 15;                 // A row / B col / D col index for this lane
    int hi = lane >> 4;
    int kb = hi * 2;                    // K sub-index within 4-wide step

    // preload all A fragments for this wave's 16x64 tile (32 VGPRs)
    v2f afr[16];
    const float* hrow = h2 + (size_t)(row0 + m) * CIO + kb;
#pragma unroll
    for (int s = 0; s < 16; ++s)
      afr[s] = *(const v2f*)(hrow + 4 * s);

    // 4 accumulators (one per 16-wide N tile), bias preloaded into C
    v8f acc[4];
#pragma unroll
    for (int t = 0; t < 4; ++t) {
      float bj = bias[t * 16 + m];
#pragma unroll
      for (int v = 0; v < 8; ++v) acc[t][v] = bj;
    }

    const float* wbase = &Wl[(size_t)m * CIO + kb];
#pragma unroll
    for (int s = 0; s < 16; ++s) {
#pragma unroll
      for (int t = 0; t < 4; ++t) {     // independent chains back-to-back
        v2f bfr = *(const v2f*)(wbase + (size_t)t * 16 * CIO + 4 * s);
        acc[t] = __builtin_amdgcn_wmma_f32_16x16x4_f32(
                     false, afr[s], false, bfr, (short)0, acc[t], false, false);
      }
    }

    // PReLU, store, per-channel partial BN stats
#pragma unroll
    for (int t = 0; t < 4; ++t) {
      int j = t * 16 + m;
      float aj = slope[j];
      float ps = 0.0f, pq = 0.0f;
#pragma unroll
      for (int v = 0; v < 8; ++v) {
        float val = acc[t][v];
        val = (val >= 0.0f) ? val : aj * val;
        h3[(size_t)(row0 + v + 8 * hi) * CIO + j] = val;
        ps += val;
        pq += val * val;
      }
      atomicAdd(&sSum[j], ps);          // ds_add_f32
      atomicAdd(&sSq[j],  pq);
    }
  }
  __syncthreads();
  if (tid < CIO) {
    atomicAddF(&chanSum[tid], sSum[tid]);
    atomicAddF(&chanSq[tid],  sSq[tid]);
  }
}

// ---------------- BN finalize: scale/shift per channel ----------------
__global__ void bnfin_kernel(const float* __restrict__ chanSum, const float* __restrict__ chanSq,
                             const float* __restrict__ gamma, const float* __restrict__ beta,
                             float* __restrict__ scale, float* __restrict__ shift, int n) {
  int c = threadIdx.x;
  if (c < CIO) {
    float inv = 1.0f / (float)n;
    float mu  = chanSum[c] * inv;
    float var = chanSq[c] * inv - mu * mu;
    float sc  = gamma[c] * rsqrtf(var + BN_EPS);
    scale[c] = sc;
    shift[c] = beta[c] - mu * sc;
  }
}

// ---------------- BN apply + mean/max pooling ----------------
// 256 threads: channel = tid&63, node group = tid>>6; 16 nodes per block.
// batch is sorted -> run-length accumulate before flushing atomics.
__global__ void pool_kernel(const float* __restrict__ h3, const int* __restrict__ batch,
                            const float* __restrict__ scale, const float* __restrict__ shift,
                            float* __restrict__ meanAcc, float* __restrict__ maxAcc,
                            float* __restrict__ cnt, int n) {
  int tid = threadIdx.x;
  int c   = tid & 63;
  int sub = tid >> 6;
  int base = blockIdx.x * 16 + sub * 4;
  float sc = scale[c], sh = shift[c];
  int   curg = -1, cn = 0;
  float sum = 0.0f, mx = 0.0f;
  for (int i = 0; i < 4; ++i) {
    int node = base + i;
    if (node >= n) break;
    int g = batch[node];
    float val = h3[(size_t)node * CIO + c] * sc + sh;
    if (g != curg) {
      if (curg >= 0) {
        atomicAddF(&meanAcc[curg * CIO + c], sum);
        atomicMaxF(&maxAcc[curg * CIO + c], mx);
        if (c == 0) atomicAddF(&cnt[curg], (float)cn);
      }
      curg = g; sum = 0.0f; mx = -__builtin_inff(); cn = 0;
    }
    sum += val;
    mx = fmaxf(mx, val);
    ++cn;
  }
  if (curg >= 0) {
    atomicAddF(&meanAcc[curg * CIO + c], sum);
    atomicMaxF(&maxAcc[curg * CIO + c], mx);
    if (c == 0) atomicAddF(&cnt[curg], (float)cn);
  }
}

// ---------------- write [G, 128] = concat(mean, max) ----------------
__global__ void writeout_kernel(const float* __restrict__ meanAcc, const float* __restrict__ maxAcc,
                                const float* __restrict__ cnt, float* __restrict__ out, int G) {
  int idx = blockIdx.x * blockDim.x + threadIdx.x;
  if (idx >= G * 128) return;
  int g = idx >> 7, c = idx & 127;
  out[idx] = (c < 64) ? meanAcc[g * 64 + c] / fmaxf(cnt[g], 1.0f)
                      : maxAcc[g * 64 + (c - 64)];
}

// ---------------- host ----------------
static inline int cdiv(long long a, long long b) { return (int)((a + b - 1) / b); }

extern "C" void kernel_launch(void* const* d_in, const int* in_sizes, int n_in,
                              void* d_out, int out_size, void* d_ws, size_t ws_size,
                              hipStream_t stream) {
  const float* x     = (const float*)d_in[0];
  const int*   ei    = (const int*)  d_in[1];
  const float* ew    = (const float*)d_in[2];
  const int*   batch = (const int*)  d_in[3];
  const float* W     = (const float*)d_in[4];
  const float* b     = (const float*)d_in[5];
  const float* a     = (const float*)d_in[6];
  const float* gamma = (const float*)d_in[7];
  const float* beta  = (const float*)d_in[8];

  const int E = in_sizes[2];
  const int N = in_sizes[3];
  const int G = out_size / 128;

  const int* rowA = ei;       // edge_index[0] = source
  const int* colA = ei + E;   // edge_index[1] = target

  // workspace layout (floats); deg doubles as dinv, h1 doubles as h3
  float* ws = (float*)d_ws;
  size_t off = 0;
  float* deg     = ws + off; off += N;
  float* h1      = ws + off; off += (size_t)N * CIO;   // hop1 out, later h3
  float* h2      = ws + off; off += (size_t)N * CIO;   // hop2 out
  float* chanSum = ws + off; off += CIO;
  float* chanSq  = ws + off; off += CIO;
  float* meanAcc = ws + off; off += (size_t)G * CIO;
  float* cnt     = ws + off; off += G;
  const long long zeroCount = (long long)off;          // contiguous zero region
  float* scaleA  = ws + off; off += CIO;
  float* shiftA  = ws + off; off += CIO;
  float* maxAcc  = ws + off; off += (size_t)G * CIO;

  const int T = 256;

  // init accumulators (every call: harness does not re-poison between replays)
  fill_kernel<<<cdiv(zeroCount, T), T, 0, stream>>>(ws, 0.0f, zeroCount);
  fill_kernel<<<cdiv((long long)G * CIO, T), T, 0, stream>>>(maxAcc, -__builtin_inff(),
                                                             (long long)G * CIO);

  // gcn_norm degrees and D^{-1/2}
  degree_kernel<<<cdiv((long long)E + N, T), T, 0, stream>>>(colA, ew, deg, E, N);
  rsqrt_kernel<<<cdiv(N, T), T, 0, stream>>>(deg, N);

  // two propagation hops (one wave per edge incl. self-loops)
  long long hopThreads = ((long long)E + N) * 32;
  hop_kernel<<<cdiv(hopThreads, T), T, 0, stream>>>(rowA, colA, ew, deg, x,  h1, E, N);
  hop_kernel<<<cdiv(hopThreads, T), T, 0, stream>>>(rowA, colA, ew, deg, h1, h2, E, N);

  // linear + PReLU + BN stats (WMMA f32 16x16x4); h3 aliases h1 (h1 fully consumed)
  gemm_bn_kernel<<<cdiv(N, 128), T, 0, stream>>>(h2, W, b, a, h1, chanSum, chanSq, N);

  // BN finalize, then apply + pool, then emit [G, 2*C]
  bnfin_kernel<<<1, 64, 0, stream>>>(chanSum, chanSq, gamma, beta, scaleA, shiftA, N);
  pool_kernel<<<cdiv(N, 16), T, 0, stream>>>(h1, batch, scaleA, shiftA, meanAcc, maxAcc, cnt, N);
  writeout_kernel<<<cdiv((long long)G * 128, T), T, 0, stream>>>(meanAcc, maxAcc, cnt,
                                                                 (float*)d_out, G);
}